// FQCE_8933531975809
// MI455X (gfx1250) — compile-verified
//
#include <hip/hip_runtime.h>
#include <math.h>

typedef __attribute__((ext_vector_type(2))) float v2f;
typedef __attribute__((ext_vector_type(8))) float v8f;

// General SU(2) rotation from the reference's _rot(phi, theta, omega):
// m00 = e^{-i(phi+omega)/2} c ; m01 = -e^{i(phi-omega)/2} s
// m10 = e^{-i(phi-omega)/2} s ; m11 =  e^{i(phi+omega)/2} c
__device__ __forceinline__ void rot_mat(float phi, float theta, float omega,
                                        float ur[2][2], float ui[2][2]) {
  float ct, st, cp, sp, cm, sm;
  __sincosf(0.5f * theta, &st, &ct);
  __sincosf(0.5f * (phi + omega), &sp, &cp);
  __sincosf(0.5f * (phi - omega), &sm, &cm);
  ur[0][0] =  cp * ct;  ui[0][0] = -sp * ct;
  ur[0][1] = -cm * st;  ui[0][1] = -sm * st;
  ur[1][0] =  cm * st;  ui[1][0] = -sm * st;
  ur[1][1] =  cp * ct;  ui[1][1] =  sp * ct;
}

// new_j = sum_i U[j][i] * old_i  (2x2 complex matvec on an amplitude pair)
__device__ __forceinline__ void apply2(float& r0, float& i0, float& r1, float& i1,
                                       const float ur[2][2], const float ui[2][2]) {
  float nr0 = ur[0][0]*r0 - ui[0][0]*i0 + ur[0][1]*r1 - ui[0][1]*i1;
  float ni0 = ur[0][0]*i0 + ui[0][0]*r0 + ur[0][1]*i1 + ui[0][1]*r1;
  float nr1 = ur[1][0]*r0 - ui[1][0]*i0 + ur[1][1]*r1 - ui[1][1]*i1;
  float ni1 = ur[1][0]*i0 + ui[1][0]*r0 + ur[1][1]*i1 + ui[1][1]*r1;
  r0 = nr0; i0 = ni0; r1 = nr1; i1 = ni1;
}

// One circuit block: params p laid out [4][6][3] (block, qubit, {phi,theta,omega}).
// Qubit q has flattened-index bit weight 2^(5-q). Fully unrolled so the 64-entry
// state arrays live in VGPRs.
__device__ __forceinline__ void circuit_block(float sr[64], float si[64],
                                              const float* __restrict__ p) {
  // Layer 0: single-qubit rotations on every qubit.
  #pragma unroll
  for (int q = 0; q < 6; ++q) {
    float ur[2][2], ui[2][2];
    rot_mat(p[q*3 + 0], p[q*3 + 1], p[q*3 + 2], ur, ui);
    const int st = 1 << (5 - q);
    #pragma unroll
    for (int idx = 0; idx < 64; ++idx) {
      if (!(idx & st)) {
        apply2(sr[idx], si[idx], sr[idx | st], si[idx | st], ur, ui);
      }
    }
  }
  // Layers 1..3: controlled rotations, control q, target (q+off)%6.
  #pragma unroll
  for (int off = 1; off <= 3; ++off) {
    #pragma unroll
    for (int q = 0; q < 6; ++q) {
      const int t = (q + off) % 6;
      float ur[2][2], ui[2][2];
      const float* pp = p + (off * 6 + q) * 3;
      rot_mat(pp[0], pp[1], pp[2], ur, ui);
      const int sc  = 1 << (5 - q);
      const int stt = 1 << (5 - t);
      #pragma unroll
      for (int idx = 0; idx < 64; ++idx) {
        if ((idx & sc) && !(idx & stt)) {
          apply2(sr[idx], si[idx], sr[idx | stt], si[idx | stt], ur, ui);
        }
      }
    }
  }
}

// Kernel A: state_e = circuit_block(uniform, entity_params[e]) for all entities.
// Row layout in ws: 128 floats, interleaved [re0, im0, re1, im1, ...].
__global__ void fqce_entity_states(const float* __restrict__ entity_params,
                                   float* __restrict__ ent_states, int E) {
  int e = blockIdx.x * blockDim.x + threadIdx.x;
  if (e >= E) return;
  float sr[64], si[64];
  #pragma unroll
  for (int i = 0; i < 64; ++i) { sr[i] = 0.125f; si[i] = 0.0f; }  // 2^{-Q/2}
  circuit_block(sr, si, entity_params + (size_t)e * 72);
  float2* dst = (float2*)(ent_states + (size_t)e * 128);
  #pragma unroll
  for (int i = 0; i < 64; ++i) dst[i] = make_float2(sr[i], si[i]);
}

#define ROWS_PER_BLOCK 64
#define LDS_STRIDE 129  // floats; odd stride -> conflict-free row writes

// Kernel B: per batch element: sp = U_rel(p) * state_s (in registers), stage sp
// rows in LDS; then per wave, two 16-row groups compute diag(SP * O^T) with
// chained V_WMMA_F32_16X16X4_F32, O rows gathered from the entity-state table.
__global__ void __launch_bounds__(ROWS_PER_BLOCK)
fqce_main(const float* __restrict__ relation_params,
          const float* __restrict__ ent_states,
          const int* __restrict__ s_idx, const int* __restrict__ p_idx,
          const int* __restrict__ o_idx, float* __restrict__ out) {
  __shared__ float sp_rows[ROWS_PER_BLOCK * LDS_STRIDE];

  const int tid = threadIdx.x;
  const int b   = blockIdx.x * ROWS_PER_BLOCK + tid;

  // ---- Phase 1: relation block applied to state_s, one element per thread ----
  {
    float sr[64], si[64];
    const float2* src = (const float2*)(ent_states + (size_t)s_idx[b] * 128);
    #pragma unroll
    for (int i = 0; i < 64; ++i) { float2 v = src[i]; sr[i] = v.x; si[i] = v.y; }
    circuit_block(sr, si, relation_params + (size_t)p_idx[b] * 72);
    float* row = &sp_rows[tid * LDS_STRIDE];
    #pragma unroll
    for (int i = 0; i < 64; ++i) { row[2*i] = sr[i]; row[2*i + 1] = si[i]; }
  }
  __syncthreads();

  // ---- Phase 2: WMMA dot products. score_b = sum_col SP[b][col] * O[b][col] ----
  const int lane = tid & 31;
  const int wave = tid >> 5;
  const int m    = lane & 15;   // row within the 16-row group (A) / col N (B)
  const int half = lane >> 4;   // selects K pair {0,1} vs {2,3}

  #pragma unroll
  for (int g = 0; g < 2; ++g) {
    const int rbase = wave * 32 + g * 16;          // first row of this group
    const int bglob = blockIdx.x * ROWS_PER_BLOCK + rbase + m;
    const float* orow = ent_states + (size_t)o_idx[bglob] * 128;
    const float* srow = &sp_rows[(rbase + m) * LDS_STRIDE];

    v8f acc = {};
    #pragma unroll
    for (int k4 = 0; k4 < 32; ++k4) {
      const int k = k4 * 4 + half * 2;
      // A (16x4 f32): vgpr0 = Row[m][K= 0|2], vgpr1 = Row[m][K= 1|3]
      v2f a;  a[0] = srow[k];  a[1] = srow[k + 1];
      // B (4x16 f32): same striping with lane index = column N
      const float2 ob = *(const float2*)(orow + k);   // 8B aligned (k even)
      v2f bv; bv[0] = ob.x; bv[1] = ob.y;
      acc = __builtin_amdgcn_wmma_f32_16x16x4_f32(
          /*neg_a=*/false, a, /*neg_b=*/false, bv,
          /*c_mod=*/(short)0, acc, /*reuse_a=*/false, /*reuse_b=*/false);
    }

    // Diagonal of C (16x16 f32, 8 VGPRs): b<8 -> vgpr b lane b; b>=8 -> vgpr b-8 lane b+16.
    const int obase = blockIdx.x * ROWS_PER_BLOCK + rbase;
    #pragma unroll
    for (int r = 0; r < 8; ++r) {
      if (lane == r)      out[obase + r]     = acc[r];
      if (lane == r + 24) out[obase + r + 8] = acc[r];
    }
  }
}

extern "C" void kernel_launch(void* const* d_in, const int* in_sizes, int n_in,
                              void* d_out, int out_size, void* d_ws, size_t ws_size,
                              hipStream_t stream) {
  const float* entity_params   = (const float*)d_in[0];
  const float* relation_params = (const float*)d_in[1];
  const int*   s_idx           = (const int*)d_in[2];
  const int*   p_idx           = (const int*)d_in[3];
  const int*   o_idx           = (const int*)d_in[4];
  float*       out             = (float*)d_out;

  const int E = in_sizes[0] / 72;   // (E,4,6,3)
  const int B = in_sizes[2];

  float* ent_states = (float*)d_ws;  // E * 128 floats (~5.1 MB for E=10000)

  fqce_entity_states<<<(E + 127) / 128, 128, 0, stream>>>(entity_params, ent_states, E);
  fqce_main<<<B / ROWS_PER_BLOCK, ROWS_PER_BLOCK, 0, stream>>>(
      relation_params, ent_states, s_idx, p_idx, o_idx, out);
}